// Twist2Mat_1949915152782
// MI455X (gfx1250) — compile-verified
//
#include <hip/hip_runtime.h>
#include <cstdint>

#define THREADS 256
#define ITEMS_PER_BLOCK 1024   // 4 items per thread, 8 wave32s per block

// ---- CDNA5 async global<->LDS helpers (gfx1250) -----------------------------
// The low 32 bits of a flat pointer into LDS are the wave-relative LDS byte
// offset (flat aperture layout: addr[31:0] = LDS offset), which is exactly
// what the async-LDS instructions take in their LDS-address VGPR.
__device__ __forceinline__ unsigned lds_off(const void* p) {
    return (unsigned)(uintptr_t)p;
}

// Streaming kernel: input read once, output written once, total traffic equals
// L2 capacity -> non-temporal hints keep this stream from churning the 192 MB L2.
__device__ __forceinline__ void async_load_b128(unsigned lds_byte, unsigned long long gaddr) {
    asm volatile("global_load_async_to_lds_b128 %0, %1, off th:TH_LOAD_NT"
                 :: "v"(lds_byte), "v"(gaddr) : "memory");
}
__device__ __forceinline__ void async_store_b128(unsigned long long gaddr, unsigned lds_byte) {
    asm volatile("global_store_async_from_lds_b128 %0, %1, off th:TH_STORE_NT"
                 :: "v"(gaddr), "v"(lds_byte) : "memory");
}
__device__ __forceinline__ void wait_async0() {
    asm volatile("s_wait_asynccnt 0" ::: "memory");
}

// ---- Rodrigues core ---------------------------------------------------------
__device__ __forceinline__ void rodrigues(float x, float y, float z, float* o) {
    float n  = sqrtf(x * x + y * y + z * z);
    float th = fmaxf(n, 1e-5f);
    float inv = 1.0f / th;
    float a0 = x * inv, a1 = y * inv, a2 = z * inv;
    float s, c;
    sincosf(th, &s, &c);   // precise ocml path; VALU is free under the HBM roofline
    float cm = 1.0f - c;
    float a00 = a0 * a0, a11 = a1 * a1, a22 = a2 * a2;
    float a01 = a0 * a1, a02 = a0 * a2, a12 = a1 * a2;
    // A2 = a a^T - I (same product structure as the reference einsum)
    o[0] = 1.0f - cm * (a11 + a22);
    o[1] = cm * a01 - s * a2;
    o[2] = cm * a02 + s * a1;
    o[3] = cm * a01 + s * a2;
    o[4] = 1.0f - cm * (a00 + a22);
    o[5] = cm * a12 - s * a0;
    o[6] = cm * a02 - s * a1;
    o[7] = cm * a12 + s * a0;
    o[8] = 1.0f - cm * (a00 + a11);
}

// ---- Main tiled kernel: full 1024-item blocks -------------------------------
__global__ __launch_bounds__(THREADS)
void twist2mat_tile(const float* __restrict__ in, float* __restrict__ out) {
    __shared__ float s_in [ITEMS_PER_BLOCK * 3];  // 12 KB
    __shared__ float s_out[ITEMS_PER_BLOCK * 9];  // 36 KB

    const unsigned tid = threadIdx.x;
    const unsigned long long inBase  = (unsigned long long)(uintptr_t)in
                                     + (unsigned long long)blockIdx.x * (ITEMS_PER_BLOCK * 3u) * 4u;
    const unsigned long long outBase = (unsigned long long)(uintptr_t)out
                                     + (unsigned long long)blockIdx.x * (ITEMS_PER_BLOCK * 9u) * 4u;
    const unsigned sInBase  = lds_off(s_in);
    const unsigned sOutBase = lds_off(s_out);

    // Phase 1: async DMA global -> LDS, 3 coalesced b128 per lane (12 KB/block)
#pragma unroll
    for (int k = 0; k < 3; ++k) {
        unsigned off = (tid + k * THREADS) * 16u;
        async_load_b128(sInBase + off, inBase + off);
    }
    wait_async0();
    __syncthreads();

    // Phase 2: compute 4 items per thread; stride-3 / stride-9 LDS (bank-conflict free)
#pragma unroll
    for (int q = 0; q < 4; ++q) {
        int i = (int)tid + q * THREADS;
        float x = s_in[3 * i + 0];
        float y = s_in[3 * i + 1];
        float z = s_in[3 * i + 2];
        float r[9];
        rodrigues(x, y, z, r);
#pragma unroll
        for (int c = 0; c < 9; ++c) s_out[9 * i + c] = r[c];
    }
    __syncthreads();

    // Phase 3: async DMA LDS -> global, 9 coalesced b128 per lane (36 KB/block)
#pragma unroll
    for (int k = 0; k < 9; ++k) {
        unsigned off = (tid + k * THREADS) * 16u;
        async_store_b128(outBase + off, sOutBase + off);
    }
    wait_async0();  // s_endpgm also performs an implicit wait-idle
}

// ---- Tail kernel (not launched for B = 4,194,304, kept for generality) ------
__global__ void twist2mat_scalar(const float* __restrict__ in, float* __restrict__ out,
                                 long long start, long long n) {
    long long i = start + (long long)blockIdx.x * blockDim.x + threadIdx.x;
    if (i >= n) return;
    float r[9];
    rodrigues(in[3 * i], in[3 * i + 1], in[3 * i + 2], r);
#pragma unroll
    for (int c = 0; c < 9; ++c) out[9 * i + c] = r[c];
}

extern "C" void kernel_launch(void* const* d_in, const int* in_sizes, int n_in,
                              void* d_out, int out_size, void* d_ws, size_t ws_size,
                              hipStream_t stream) {
    const float* in = (const float*)d_in[0];
    float* out = (float*)d_out;
    const long long B = (long long)in_sizes[0] / 3;

    const long long full = B / ITEMS_PER_BLOCK;
    if (full > 0) {
        twist2mat_tile<<<(unsigned)full, THREADS, 0, stream>>>(in, out);
    }
    const long long rem = B - full * ITEMS_PER_BLOCK;
    if (rem > 0) {
        long long start = full * (long long)ITEMS_PER_BLOCK;
        twist2mat_scalar<<<(unsigned)((rem + THREADS - 1) / THREADS), THREADS, 0, stream>>>(
            in, out, start, B);
    }
}